// LabelStatsAccumulatorModule_9732395893064
// MI455X (gfx1250) — compile-verified
//
#include <hip/hip_runtime.h>
#include <hip/hip_bf16.h>
#include <math.h>

typedef __attribute__((ext_vector_type(4))) float f32x4;
typedef __attribute__((ext_vector_type(4))) int   i32x4;

// ---------------------------------------------------------------------------
// Fire-and-forget device-scope float atomics (CDNA5 GLOBAL_ATOMIC_*_NUM_F32).
// SADDR form: 32-bit byte offset in VGPR, uniform 64-bit base in SGPR pair.
// No-return variant -> tracked with STOREcnt; S_ENDPGM's implicit wait-idle
// (and the kernel boundary before the finalize pass) guarantees completion.
// ---------------------------------------------------------------------------
__device__ __forceinline__ void gatom_add_f32(float* base, int byteoff, float v) {
  asm volatile("global_atomic_add_f32 %0, %1, %2 scope:SCOPE_DEV"
               :: "v"(byteoff), "v"(v), "s"(base) : "memory");
}
__device__ __forceinline__ void gatom_min_f32(float* base, int byteoff, float v) {
  asm volatile("global_atomic_min_num_f32 %0, %1, %2 scope:SCOPE_DEV"
               :: "v"(byteoff), "v"(v), "s"(base) : "memory");
}
__device__ __forceinline__ void gatom_max_f32(float* base, int byteoff, float v) {
  asm volatile("global_atomic_max_num_f32 %0, %1, %2 scope:SCOPE_DEV"
               :: "v"(byteoff), "v"(v), "s"(base) : "memory");
}

// ---------------------------------------------------------------------------
// Pass 0: initialize the L2-resident accumulator (d_ws is poisoned by the
// harness and NOT re-initialized between timed replays).
// ---------------------------------------------------------------------------
__global__ void labelstats_init_kernel(float* __restrict__ sums,
                                       float* __restrict__ mins,
                                       float* __restrict__ maxs,
                                       int L) {
  int i = blockIdx.x * blockDim.x + threadIdx.x;
  if (i < L) {
    sums[i] = 0.0f;
    mins[i] = INFINITY;    // identity for segment_min
    maxs[i] = -INFINITY;   // identity for segment_max
  }
}

// ---------------------------------------------------------------------------
// Pass 1: stream 268 MB once with NT 128-bit loads, scatter 3 device-scope
// atomics per element into the 1.2 MB accumulator (stays resident in L2).
// ---------------------------------------------------------------------------
__global__ void labelstats_accum_kernel(const float* __restrict__ x,
                                        const int*   __restrict__ lab,
                                        float* __restrict__ sums,
                                        float* __restrict__ mins,
                                        float* __restrict__ maxs,
                                        long long n4, long long n) {
  const long long stride = (long long)gridDim.x * blockDim.x;
  const long long gid    = (long long)blockIdx.x * blockDim.x + threadIdx.x;

  const f32x4* __restrict__ x4 = (const f32x4*)x;
  const i32x4* __restrict__ l4 = (const i32x4*)lab;

  for (long long i = gid; i < n4; i += stride) {
    // Streaming data: read once, keep out of caches (th:TH_LOAD_NT).
    f32x4 xv = __builtin_nontemporal_load(x4 + i);
    i32x4 lv = __builtin_nontemporal_load(l4 + i);
#pragma unroll
    for (int k = 0; k < 4; ++k) {
      const int   off = lv[k] << 2;   // byte offset, labels < 100k fits i32
      const float v   = xv[k];
      gatom_add_f32(sums, off, v);
      gatom_min_f32(mins, off, v);
      gatom_max_f32(maxs, off, v);
    }
  }

  // Scalar tail (N is a multiple of 4 in practice, but stay correct).
  const long long base = n4 * 4;
  for (long long i = base + gid; i < n; i += stride) {
    const float v   = __builtin_nontemporal_load(x + i);
    const int   off = __builtin_nontemporal_load(lab + i) << 2;
    gatom_add_f32(sums, off, v);
    gatom_min_f32(mins, off, v);
    gatom_max_f32(maxs, off, v);
  }
}

// ---------------------------------------------------------------------------
// Pass 2: finalize -> out[l] = {sum, sum/max(count,1), min, max} as one b128
// store per label.
// ---------------------------------------------------------------------------
__global__ void labelstats_finalize_kernel(const float* __restrict__ sums,
                                           const float* __restrict__ mins,
                                           const float* __restrict__ maxs,
                                           const int*   __restrict__ cnt,
                                           float* __restrict__ out,
                                           int L) {
  int i = blockIdx.x * blockDim.x + threadIdx.x;
  if (i < L) {
    const float s = sums[i];
    int   ci = cnt[i];
    const float c = (float)(ci > 1 ? ci : 1);
    f32x4 o;
    o.x = s;
    o.y = s / c;
    o.z = mins[i];
    o.w = maxs[i];
    __builtin_nontemporal_store(o, (f32x4*)out + i);
  }
}

extern "C" void kernel_launch(void* const* d_in, const int* in_sizes, int n_in,
                              void* d_out, int out_size, void* d_ws, size_t ws_size,
                              hipStream_t stream) {
  const float* x    = (const float*)d_in[0];   // input  [N] f32
  const int*   lab  = (const int*)  d_in[1];   // labels [N] i32
  const int*   cnt  = (const int*)  d_in[2];   // labelcount [L] i32
  const long long N = (long long)in_sizes[0];
  const int       L = in_sizes[2];

  // Workspace layout: sums[L] | mins[L] | maxs[L]  (3*L*4 bytes = 1.2 MB)
  float* sums = (float*)d_ws;
  float* mins = sums + L;
  float* maxs = mins + L;
  float* out  = (float*)d_out;                 // [L,4] f32

  const int  TPB    = 256;                     // 8 wave32 per block
  const int  gridL  = (L + TPB - 1) / TPB;

  labelstats_init_kernel<<<gridL, TPB, 0, stream>>>(sums, mins, maxs, L);

  const long long n4 = N / 4;
  long long blocks64 = (n4 + TPB - 1) / TPB;
  if (blocks64 > 32768) blocks64 = 32768;      // grid-stride beyond this
  if (blocks64 < 1) blocks64 = 1;
  labelstats_accum_kernel<<<(int)blocks64, TPB, 0, stream>>>(
      x, lab, sums, mins, maxs, n4, N);

  labelstats_finalize_kernel<<<gridL, TPB, 0, stream>>>(
      sums, mins, maxs, cnt, out, L);
}